// GNN_17918603558958
// MI455X (gfx1250) — compile-verified
//
#include <hip/hip_runtime.h>

#define N_NODES 100000
#define N_EDGES 3200000
#define D_FEAT  128
#define HIDDEN  64

typedef __attribute__((ext_vector_type(2))) float v2f;
typedef __attribute__((ext_vector_type(8))) float v8f;

// ---------------------------------------------------------------------------
// init: zero accumulator [N,64] (float4 stores), deg = 1.0 (self loop),
// zero scalar sum
// ---------------------------------------------------------------------------
__global__ void gcn_init(float4* __restrict__ acc4, float* __restrict__ deg,
                         float* __restrict__ gsum) {
  int i = blockIdx.x * blockDim.x + threadIdx.x;
  if (i < (N_NODES * HIDDEN) / 4) acc4[i] = make_float4(0.f, 0.f, 0.f, 0.f);
  if (i < N_NODES) deg[i] = 1.0f;         // self-loop contribution
  if (i == 0) gsum[0] = 0.0f;
}

// ---------------------------------------------------------------------------
// degree: deg[dst] += 1 for every edge
// ---------------------------------------------------------------------------
__global__ void gcn_degree(const long long* __restrict__ ei, float* __restrict__ deg) {
  int e = blockIdx.x * blockDim.x + threadIdx.x;
  if (e >= N_EDGES) return;
  int d = (int)ei[N_EDGES + e];           // edge_index[1][e]
  atomicAdd(&deg[d], 1.0f);
}

// ---------------------------------------------------------------------------
// dinv: deg -> rsqrt(deg) in place (deg >= 1 always due to self loops)
// ---------------------------------------------------------------------------
__global__ void gcn_dinv(float* __restrict__ deg) {
  int i = blockIdx.x * blockDim.x + threadIdx.x;
  if (i < N_NODES) deg[i] = rsqrtf(deg[i]);
}

// ---------------------------------------------------------------------------
// GEMM h = x @ W_conv, full fp32 via V_WMMA_F32_16X16X4_F32.
// W_conv (32 KB) is staged to LDS with GLOBAL_LOAD_ASYNC_TO_LDS_B128
// (ASYNCcnt-tracked fire-and-forget copy; ISA 15.18.3 opcode 98), then each
// wave computes one 16-row x 64-col tile with 4 accumulators, K stepped by 4.
//
// A (16x4 f32) VGPR layout: lanes 0-15 row M=lane, v0=K0 v1=K1;
//                           lanes 16-31 row M=lane-16, v0=K2 v1=K3.
// B (4x16 f32): v0 = {K0 row | K2 row}, v1 = {K1 row | K3 row}, N = lane&15.
// C/D (16x16 f32): VGPR i -> M=i (lanes 0-15) / M=i+8 (lanes 16-31), N=lane&15.
// ---------------------------------------------------------------------------
__global__ void __launch_bounds__(256)
gcn_gemm_wmma(const float* __restrict__ x, const float* __restrict__ W,
              float* __restrict__ h) {
  __shared__ float Wlds[D_FEAT * HIDDEN];           // 32 KB of 320 KB LDS

  // async global->LDS staging of W: 2048 x 16B lane-transfers across the block
  for (int i = threadIdx.x; i < (D_FEAT * HIDDEN) / 4; i += 256) {
    unsigned lds_off = (unsigned)(size_t)(&Wlds[i * 4]);  // flat addr low bits = LDS byte offset
    const float* gp = W + i * 4;
    asm volatile("global_load_async_to_lds_b128 %0, %1, off"
                 :: "v"(lds_off), "v"(gp) : "memory");
  }
  asm volatile("s_wait_asynccnt 0x0" ::: "memory");
  __syncthreads();

  const int wave = (blockIdx.x * 256 + threadIdx.x) >> 5;   // uniform per wave
  const int lane = threadIdx.x & 31;
  if (wave >= N_NODES / 16) return;                  // wave-uniform: EXEC stays all-1s

  const int row_base = wave * 16;
  const int m  = lane & 15;                          // row / N column within tile
  const int kh = lane >> 4;                          // K-half selector
  const float* xrow = x + (size_t)(row_base + m) * D_FEAT;
  __builtin_prefetch(xrow, 0, 0);                    // global_prefetch_b8

  v8f c0 = {}, c1 = {}, c2 = {}, c3 = {};
  #pragma unroll 4
  for (int k0 = 0; k0 < D_FEAT; k0 += 4) {
    const int ka = k0 + 2 * kh;
    v2f a;
    a.x = xrow[ka];
    a.y = xrow[ka + 1];

    const float* w0 = &Wlds[ka * HIDDEN + m];
    const float* w1 = &Wlds[(ka + 1) * HIDDEN + m];
    v2f b0; b0.x = w0[0];  b0.y = w1[0];
    v2f b1; b1.x = w0[16]; b1.y = w1[16];
    v2f b2; b2.x = w0[32]; b2.y = w1[32];
    v2f b3; b3.x = w0[48]; b3.y = w1[48];

    c0 = __builtin_amdgcn_wmma_f32_16x16x4_f32(false, a, false, b0, (short)0, c0, false, false);
    c1 = __builtin_amdgcn_wmma_f32_16x16x4_f32(false, a, false, b1, (short)0, c1, false, false);
    c2 = __builtin_amdgcn_wmma_f32_16x16x4_f32(false, a, false, b2, (short)0, c2, false, false);
    c3 = __builtin_amdgcn_wmma_f32_16x16x4_f32(false, a, false, b3, (short)0, c3, false, false);
  }

  float* hp = h + (size_t)row_base * HIDDEN;
  #pragma unroll
  for (int i = 0; i < 8; ++i) {
    const int r = i + 8 * kh;
    hp[r * HIDDEN + m]      = c0[i];
    hp[r * HIDDEN + 16 + m] = c1[i];
    hp[r * HIDDEN + 32 + m] = c2[i];
    hp[r * HIDDEN + 48 + m] = c3[i];
  }
}

// ---------------------------------------------------------------------------
// scatter: one wave per edge; lane handles 2 features (float2 gather of h,
// two f32 atomic adds). acc[dst,f] += h[src,f] * dinv[src]*dinv[dst]
// ---------------------------------------------------------------------------
__global__ void __launch_bounds__(256)
gcn_scatter(const long long* __restrict__ ei, const float* __restrict__ h,
            const float* __restrict__ dinv, float* __restrict__ acc) {
  const unsigned t = blockIdx.x * 256u + threadIdx.x;
  const int e    = (int)(t >> 5);        // wave-uniform edge id
  const int lane = (int)(t & 31u);
  if (e >= N_EDGES) return;
  const int s = (int)ei[e];
  const int d = (int)ei[N_EDGES + e];
  const float w = dinv[s] * dinv[d];
  const float2 hv = *(const float2*)&h[(size_t)s * HIDDEN + 2 * lane];
  float* ap = &acc[(size_t)d * HIDDEN + 2 * lane];
  atomicAdd(ap,     hv.x * w);
  atomicAdd(ap + 1, hv.y * w);
}

// ---------------------------------------------------------------------------
// final: per node add self-loop term + bias, ReLU, dot with W_fin,
// wave-reduce, single scalar atomic accumulation
// ---------------------------------------------------------------------------
__global__ void __launch_bounds__(256)
gcn_final(const float* __restrict__ acc, const float* __restrict__ h,
          const float* __restrict__ dinv, const float* __restrict__ b_conv,
          const float* __restrict__ W_fin, float* __restrict__ gsum) {
  const int node = (blockIdx.x * 256 + threadIdx.x) >> 5;
  const int lane = threadIdx.x & 31;
  if (node >= N_NODES) return;
  const size_t base = (size_t)node * HIDDEN;
  const float di = dinv[node];
  const float self_w = di * di;

  float p = 0.0f;
  #pragma unroll
  for (int f0 = 0; f0 < HIDDEN; f0 += 32) {
    const int f = f0 + lane;
    float v = acc[base + f] + h[base + f] * self_w + b_conv[f];
    v = fmaxf(v, 0.0f);                   // ReLU
    p += v * W_fin[f];
  }
  #pragma unroll
  for (int off = 16; off > 0; off >>= 1) p += __shfl_down(p, off, 32);
  if (lane == 0) atomicAdd(gsum, p);
}

// ---------------------------------------------------------------------------
// epilogue: mean over nodes + final bias -> d_out[0]
// ---------------------------------------------------------------------------
__global__ void gcn_epilogue(const float* __restrict__ gsum,
                             const float* __restrict__ b_fin,
                             float* __restrict__ out) {
  out[0] = gsum[0] * (1.0f / (float)N_NODES) + b_fin[0];
}

extern "C" void kernel_launch(void* const* d_in, const int* in_sizes, int n_in,
                              void* d_out, int out_size, void* d_ws, size_t ws_size,
                              hipStream_t stream) {
  const float*     x      = (const float*)d_in[0];
  const long long* ei     = (const long long*)d_in[1];   // int64 [2, E]
  const float*     W_conv = (const float*)d_in[2];
  const float*     b_conv = (const float*)d_in[3];
  const float*     W_fin  = (const float*)d_in[4];
  const float*     b_fin  = (const float*)d_in[5];
  float*           out    = (float*)d_out;

  // workspace layout (floats)
  float* h    = (float*)d_ws;                        // N*64
  float* acc  = h   + (size_t)N_NODES * HIDDEN;      // N*64
  float* dinv = acc + (size_t)N_NODES * HIDDEN;      // N   (deg -> rsqrt in place)
  float* gsum = dinv + N_NODES;                      // 1

  gcn_init<<<(N_NODES * HIDDEN / 4 + 255) / 256, 256, 0, stream>>>((float4*)acc, dinv, gsum);
  gcn_degree<<<(N_EDGES + 255) / 256, 256, 0, stream>>>(ei, dinv);
  gcn_dinv<<<(N_NODES + 255) / 256, 256, 0, stream>>>(dinv);

  const int waves = N_NODES / 16;                    // 6250 tiles, exact
  gcn_gemm_wmma<<<(waves + 7) / 8, 256, 0, stream>>>(x, W_conv, h);

  const long long scat_threads = (long long)N_EDGES * 32;  // one wave per edge
  gcn_scatter<<<(unsigned)((scat_threads + 255) / 256), 256, 0, stream>>>(ei, h, dinv, acc);

  gcn_final<<<(N_NODES * 32 + 255) / 256, 256, 0, stream>>>(acc, h, dinv, b_conv, W_fin, gsum);
  gcn_epilogue<<<1, 1, 0, stream>>>(gsum, b_fin, out);
}